// CostVolume_15668040696010
// MI455X (gfx1250) — compile-verified
//
#include <hip/hip_runtime.h>

typedef __attribute__((ext_vector_type(2))) float v2f;
typedef __attribute__((ext_vector_type(8))) float v8f;

#define B_ 4
#define C_ 64
#define H_ 256
#define W_ 512
#define D_ 48
#define HW_ (H_ * W_)
#define RS 68                      // LDS row stride in floats: 68*l mod 64 = 4l -> lane l owns banks 4l..4l+3
#define PADW 48                    // zero pad rows for w' < 0 (reproduces reference zero-fill)
#define RROWS (W_ + PADW)          // 560
#define L_OFF 0
#define R_OFF (W_ * RS)            // 34816 words
#define S_OFF (R_OFF + RROWS * RS) // 72896 words
#define STG_W (D_ * 16 + 16)       // 784 words per wave: 48x16 valid + 16-float trash row
#define LDS_WORDS (S_OFF + 8 * STG_W)
#define LDS_BYTES (LDS_WORDS * 4)  // 316672 B <= 320KB WGP LDS

static_assert(LDS_BYTES <= 320 * 1024, "exceeds WGP LDS");

__global__ __launch_bounds__(256, 1)
void costvol_wmma(const float* __restrict__ L,
                  const float* __restrict__ R,
                  float* __restrict__ out)
{
    extern __shared__ float lds[];
    float* sL = lds + L_OFF;   // [512][RS]  transposed left slice  [w][c]
    float* sR = lds + R_OFF;   // [560][RS]  transposed right slice [48+w'][c], rows 0..47 zero
    float* sS = lds + S_OFF;   // [8][STG_W] per-wave output staging [d][w_local] + trash row

    const int tid  = threadIdx.x;
    const int bh   = blockIdx.x;         // one WG per (b,h): each input byte pulled from HBM once
    const int b    = bh >> 8;
    const int h    = bh & (H_ - 1);
    const int wave = tid >> 5;           // 8 wave32's
    const int lane = tid & 31;

    // zero-fill the 48-row left pad of sR (w' < 0 => product 0 => mean 0)
    for (int t = tid; t < PADW * RS; t += 256) sR[t] = 0.0f;

    // ---- async transpose stage: global [c][w] -> LDS [w][c] ----------------
    // GLOBAL_LOAD_ASYNC_TO_LDS_B32: per-lane global addr (128B coalesced) ->
    // per-lane LDS addr (transpose scatter). No VGPR staging; ASYNCcnt-tracked.
    {
        const uint64_t Lg = (uint64_t)(uintptr_t)(L + (size_t)b * C_ * HW_ + (size_t)h * W_);
        const uint64_t Rg = (uint64_t)(uintptr_t)(R + (size_t)b * C_ * HW_ + (size_t)h * W_);
        for (int t = wave; t < C_ * (W_ / 32); t += 8) {   // 64 c-rows x 16 chunks of 32 w
            const int c = t >> 4;
            const int w = ((t & 15) << 5) + lane;
            const uint64_t gl = Lg + (((uint64_t)c * HW_ + (uint64_t)w) << 2);
            const uint64_t gr = Rg + (((uint64_t)c * HW_ + (uint64_t)w) << 2);
            const uint32_t dl = (uint32_t)((w * RS + c) << 2);
            const uint32_t dr = (uint32_t)((R_OFF + (PADW + w) * RS + c) << 2);
            asm volatile("global_load_async_to_lds_b32 %0, %1, off"
                         :: "v"(dl), "v"(gl) : "memory");
            asm volatile("global_load_async_to_lds_b32 %0, %1, off"
                         :: "v"(dr), "v"(gr) : "memory");
        }
    }
    asm volatile("s_wait_asynccnt 0" ::: "memory");   // our async writes landed in LDS
    __syncthreads();                                   // (barrier covers the ds zero-fill)

    // ---- banded-Gram WMMA compute ------------------------------------------
    const int lrow = lane & 15;          // M for A-tile / N for B-tile
    const int koff = (lane >> 4) * 2;    // lanes<16 -> K{0,1}, lanes>=16 -> K{2,3}

    float* stg = sS + wave * STG_W;      // wave-private staging: no cross-wave sync

    for (int i = wave; i < W_ / 16; i += 8) {
        const int w0 = i * 16;

        // A tile (16x4 fp32 per K-step): lane l<16 holds M=l,K{0,1}; l>=16 holds M=l-16,K{2,3}
        v2f a[16];
        const int abase = (w0 + lrow) * RS + koff;
        #pragma unroll
        for (int kk = 0; kk < 16; ++kk)
            a[kk] = *(const v2f*)(sL + abase + kk * 4);

        #pragma unroll
        for (int dlt = 0; dlt < 4; ++dlt) {            // band offsets: d = 16*dlt + m - n
            const int w0p   = w0 - 16 * dlt;            // w' tile origin (may reach zero pad)
            const int rbase = (PADW + w0p + lrow) * RS + koff;
            v8f acc = {0.f, 0.f, 0.f, 0.f, 0.f, 0.f, 0.f, 0.f};
            #pragma unroll
            for (int kk = 0; kk < 16; ++kk) {           // K = 64 in 16 steps of 4
                v2f bv = *(const v2f*)(sR + rbase + kk * 4);
                acc = __builtin_amdgcn_wmma_f32_16x16x4_f32(
                        false, a[kk], false, bv, (short)0, acc, false, false);
            }
            // C layout: VGPR v, lanes 0-15: M=v, N=lane; lanes 16-31: M=v+8, N=lane-16.
            // Branchless diagonal scatter: invalid d -> trash row (cndmask, no exec branch).
            const int n   = lrow;
            const int mhi = (lane >> 4) * 8;
            #pragma unroll
            for (int v = 0; v < 8; ++v) {
                const int m    = v + mhi;
                const int d    = 16 * dlt + m - n;       // each valid (d,w) produced once
                const int base = ((unsigned)d < (unsigned)D_) ? d * 16 : D_ * 16;
                stg[base + m] = acc[v] * (1.0f / (float)C_);
            }
        }

        asm volatile("s_wait_dscnt 0" ::: "memory");     // staging RAW guard (same wave)

        // coalesced writeback: 48 d-rows x 16 floats (64B rows) for this w-tile
        #pragma unroll
        for (int e = 0; e < 6; ++e) {
            const int flat = e * 128 + lane * 4;
            const int d  = flat >> 4;
            const int wl = flat & 15;
            float4 val = *(const float4*)(stg + flat);
            float* op = out + (((size_t)(b * D_ + d) * H_ + h) * W_ + w0 + wl);
            *(float4*)op = val;
        }
    }
}

extern "C" void kernel_launch(void* const* d_in, const int* in_sizes, int n_in,
                              void* d_out, int out_size, void* d_ws, size_t ws_size,
                              hipStream_t stream) {
    const float* L = (const float*)d_in[0];
    const float* R = (const float*)d_in[1];
    // d_in[2] = max_disp (structurally fixed at 48)
    float* out = (float*)d_out;

    hipFuncSetAttribute((const void*)costvol_wmma,
                        hipFuncAttributeMaxDynamicSharedMemorySize, LDS_BYTES);

    dim3 grid(B_ * H_);
    dim3 block(256);
    costvol_wmma<<<grid, block, LDS_BYTES, stream>>>(L, R, out);
}